// AVIDSimilarityPositiveExpansion_69458211111272
// MI455X (gfx1250) — compile-verified
//
#include <hip/hip_runtime.h>

typedef float v2f __attribute__((ext_vector_type(2)));
typedef float v8f __attribute__((ext_vector_type(8)));
typedef int   b128_t __attribute__((vector_size(16)));   // matches builtin param

#define T_INV   (1.0f / 0.07f)
#define EPSN    1e-12f
#define DIM     128
#define BS      256
#define NPOS    8
#define NNEG    1024
#define NKEY    (1 + NPOS + NNEG)     /* 1033 */
#define TILES   ((NKEY + 15) / 16)    /* 65  */

#if __has_builtin(__builtin_amdgcn_global_load_async_to_lds_b128)
#define HAVE_ASYNC_LDS 1
#else
#define HAVE_ASYNC_LDS 0
#endif

typedef __attribute__((address_space(1))) b128_t* gptr128;
typedef __attribute__((address_space(3))) b128_t* lptr128;

__device__ __forceinline__ void wait_async0() {
#if __has_builtin(__builtin_amdgcn_s_wait_asynccnt)
    __builtin_amdgcn_s_wait_asynccnt(0);
#else
    asm volatile("s_wait_asynccnt 0" ::: "memory");
#endif
}

// Kernel 1: normalize queries, fold in 1/T.  One wave per (batch, dir).
// dir 0 -> video query (scored against view2/audio memory)
// dir 1 -> audio query (scored against view1/video memory)
__global__ void avid_normalize_q(const float* __restrict__ video,
                                 const float* __restrict__ audio,
                                 float* __restrict__ qout) {
    const int b    = blockIdx.x;
    const int dir  = blockIdx.y;
    const int lane = threadIdx.x;   // 0..31, wave32
    const float* src = (dir == 0) ? video : audio;

    float4 v = ((const float4*)(src + b * DIM))[lane];
    float s = v.x * v.x + v.y * v.y + v.z * v.z + v.w * v.w;
#pragma unroll
    for (int off = 16; off > 0; off >>= 1)
        s += __shfl_xor(s, off, 32);

    const float scale = T_INV / fmaxf(sqrtf(s), EPSN);
    float4 o;
    o.x = v.x * scale; o.y = v.y * scale; o.z = v.z * scale; o.w = v.w * scale;
    ((float4*)(qout + (dir * BS + b) * DIM))[lane] = o;
}

// Kernel 2: one wave per (16-key tile, batch, dir).
// Gather 16 rows (512B each, fully coalesced) into LDS — via async
// direct-to-LDS copies when available — then 32 chained
// V_WMMA_F32_16X16X4_F32 accumulate the 128-dim dot products in fp32.
__global__ void avid_score_wmma(const float* __restrict__ view1,
                                const float* __restrict__ view2,
                                const int*   __restrict__ y,
                                const int*   __restrict__ pos_idx,
                                const int*   __restrict__ neg_idx,
                                const float* __restrict__ qnorm,
                                float*       __restrict__ out) {
    __shared__ __align__(16) float ldsA[16 * DIM];  // 16 gathered rows
    __shared__ __align__(16) float ldsQ[DIM];       // scaled query

    const int tile = blockIdx.x;
    const int b    = blockIdx.y;
    const int dir  = blockIdx.z;
    const int lane = threadIdx.x;     // 0..31
    const int m    = lane & 15;       // row within tile
    const int hi   = lane >> 4;       // half-wave selector (K 0/1 vs 2/3)

    const float* mem = (dir == 0) ? view2 : view1;
    const int k0 = tile * 16;

    // Key index for row m of this tile (lanes 16-31 mirror lanes 0-15).
    const int k = k0 + m;
    int idx;
    if (k == 0 || k >= NKEY) idx = y[b];
    else if (k <= NPOS)      idx = pos_idx[b * NPOS + (k - 1)];
    else                     idx = neg_idx[b * NNEG + (k - NPOS - 1)];

    const float* qsrc = qnorm + (dir * BS + b) * DIM;

#if HAVE_ASYNC_LDS
    // Async gather: memory -> LDS directly, no VGPR staging.
    __builtin_amdgcn_global_load_async_to_lds_b128(
        (gptr128)(qsrc + lane * 4), (lptr128)(ldsQ + lane * 4), 0, 0);
#pragma unroll
    for (int r = 0; r < 16; ++r) {
        const int ridx = __shfl(idx, r, 32);
        __builtin_amdgcn_global_load_async_to_lds_b128(
            (gptr128)(mem + (size_t)ridx * DIM + lane * 4),
            (lptr128)(ldsA + r * DIM + lane * 4), 0, 0);
    }
    wait_async0();
    asm volatile("" ::: "memory");  // don't hoist LDS reads above the wait
#else
    // Fallback: load to VGPRs, store to LDS.
    ((float4*)ldsQ)[lane] = ((const float4*)qsrc)[lane];
#pragma unroll
    for (int r = 0; r < 16; ++r) {
        const int ridx = __shfl(idx, r, 32);
        const float4 rv = ((const float4*)(mem + (size_t)ridx * DIM))[lane];
        ((float4*)(ldsA + r * DIM))[lane] = rv;
    }
    __syncthreads();
#endif

    // A fragment layout (16x4 f32): lanes 0-15 hold row m, K=0..1;
    // lanes 16-31 hold row m, K=2..3.  B replicates q across all N columns.
    v8f acc = {};
    const char* aBase = (const char*)ldsA + m * (DIM * 4) + hi * 8;
    const char* qBase = (const char*)ldsQ + hi * 8;
#pragma unroll
    for (int c = 0; c < 32; ++c) {
        v2f afrag = *(const v2f*)(aBase + c * 16);
        v2f bfrag = *(const v2f*)(qBase + c * 16);
        acc = __builtin_amdgcn_wmma_f32_16x16x4_f32(
            /*neg_a=*/false, afrag, /*neg_b=*/false, bfrag,
            /*c_mod=*/(short)0, acc, /*reuse_a=*/false, /*reuse_b=*/false);
    }

    // D layout: lane 0 holds D[0..7][0], lane 16 holds D[8..15][0].
    if (m == 0) {
        const int kbase = k0 + hi * 8;
        float* o = out + ((size_t)dir * BS + b) * NKEY + kbase;
#pragma unroll
        for (int i = 0; i < 8; ++i) {
            if (kbase + i < NKEY) o[i] = acc[i];
        }
    }
}

extern "C" void kernel_launch(void* const* d_in, const int* in_sizes, int n_in,
                              void* d_out, int out_size, void* d_ws, size_t ws_size,
                              hipStream_t stream) {
    const float* video = (const float*)d_in[0];
    const float* audio = (const float*)d_in[1];
    const float* view1 = (const float*)d_in[2];
    const float* view2 = (const float*)d_in[3];
    const int*   yidx  = (const int*)d_in[4];
    const int*   pidx  = (const int*)d_in[5];
    const int*   nidx  = (const int*)d_in[6];
    float* out   = (float*)d_out;
    float* qnorm = (float*)d_ws;   // 2*256*128 floats = 256 KB

    avid_normalize_q<<<dim3(BS, 2), 32, 0, stream>>>(video, audio, qnorm);
    avid_score_wmma<<<dim3(TILES, BS, 2), 32, 0, stream>>>(
        view1, view2, yidx, pidx, nidx, qnorm, out);
}